// SelfAttention_31954556682766
// MI455X (gfx1250) — compile-verified
//
#include <hip/hip_runtime.h>
#include <hip/hip_bf16.h>
#include <math.h>

#define C_ 8
#define T_ 4096
#define E_ 1024
#define H_ 16
#define D_ 64
#define SPLIT_ 8            // workgroups per (c,h)
#define TPW_ 4              // 16-row tiles per wave
// per (c,h): T_/16 = 256 tiles = SPLIT_ workgroups * 8 waves * TPW_ tiles

typedef __attribute__((ext_vector_type(2))) float v2f;
typedef __attribute__((ext_vector_type(4))) float v4f;
typedef __attribute__((ext_vector_type(8))) float v8f;
typedef __attribute__((ext_vector_type(4))) unsigned int u32x4;
typedef __attribute__((ext_vector_type(8))) int i32x8;
typedef __attribute__((ext_vector_type(4))) int i32x4;

#define MEMBAR() asm volatile("" ::: "memory")

// ---------------- kernel 0: MT[b*64+a] = M[a][b] = sum_e Wq[e,a]*Wk[e,b] ----
__global__ __launch_bounds__(256)
void k0_prep(const float* __restrict__ Wq, const float* __restrict__ Wk,
             float* __restrict__ mt) {
    __shared__ float lq[4096];
    __shared__ float lk[4096];
    int tid = threadIdx.x;
    for (int i = tid; i < 4096; i += 256) { lq[i] = Wq[i]; lk[i] = Wk[i]; }
    __syncthreads();
    for (int i = tid; i < 4096; i += 256) {
        int a = i & 63, b = i >> 6;
        float acc = 0.f;
        #pragma unroll 8
        for (int e = 0; e < 64; ++e) acc += lq[e * 64 + a] * lk[e * 64 + b];
        mt[b * 64 + a] = acc;        // transposed store
    }
}

// Build the tensor-DMA descriptor for one 16x64 f32 tile (row stride E_ elems)
// and issue TENSOR_LOAD_TO_LDS.  LDS rows land padded to 68 dwords.
__device__ __forceinline__ void tdm_load_tile(const float* gsrc, unsigned lds_byte_addr) {
    unsigned long long ga = (unsigned long long)(uintptr_t)gsrc;
    u32x4 g0;
    g0[0] = 1u;                                          // count=1, user descriptor
    g0[1] = lds_byte_addr;                               // LDS dest (bytes)
    g0[2] = (unsigned)(ga & 0xFFFFFFFFu);                // global_addr[31:0]
    g0[3] = (unsigned)((ga >> 32) & 0x1FFFFFFu) | (2u << 30);  // addr[56:32] | type=2
    i32x8 g1;
    g1[0] = (2 << 16)        // data_size = 4 bytes
          | (1 << 20)        // pad_enable
          | (5 << 22)        // pad_interval: 64 dwords
          | (3 << 25);       // pad_amount : 4 dwords  -> row stride 68 dwords
    g1[1] = (64 << 16);      // tensor_dim0[15:0] = 64 (bits 63:48)
    g1[2] = (16 << 16);      // tensor_dim1[15:0] = 16 (bits 95:80)
    g1[3] = (64 << 16);      // tile_dim0 = 64        (bits 127:112)
    g1[4] = 16;              // tile_dim1 = 16        (bits 143:128)
    g1[5] = E_;              // tensor_dim0_stride = 1024 elements
    g1[6] = 0;
    g1[7] = 0;
    i32x4 z4 = {0, 0, 0, 0};
    i32x8 z8 = {0, 0, 0, 0, 0, 0, 0, 0};
    __builtin_amdgcn_tensor_load_to_lds(g0, g1, z4, z4, z8, 0);
}

// ---------------- kernel 1: diag scores via f32 WMMA + TDM staging ----------
__global__ __launch_bounds__(256)
void k1_scores(const float* __restrict__ x, const float* __restrict__ mt,
               float* __restrict__ sbuf, float* __restrict__ pxs) {
    __shared__ float lmt[64 * 68];          // M^T, stride 68 (conflict-free b64)
    __shared__ float lqs[8 * 16 * 68];      // per-wave Q relayout scratch
    __shared__ float lut[8 * 2 * 16 * 68];  // per-wave double-buffered U tiles (TDM dest)

    const int tid = threadIdx.x;
    const int bid = blockIdx.x;
    const int c  = bid / (H_ * SPLIT_);
    const int h  = (bid / SPLIT_) % H_;
    const int sp = bid % SPLIT_;

    for (int i = tid; i < 4096; i += 256)
        lmt[(i >> 6) * 68 + (i & 63)] = mt[i];
    __syncthreads();

    const int w    = tid >> 5;   // wave in workgroup
    const int l    = tid & 31;   // lane
    const int row  = l & 15;     // M index (A frags) / N index (B frags)
    const int half = l >> 4;     // K-half select

    float* qs = &lqs[w * (16 * 68)];
    float* ut = &lut[w * (2 * 16 * 68)];
    const unsigned ut_byte0 = (unsigned)(uintptr_t)(void*)ut;
    const int gw = bid * 8 + w;          // global wave id
    const int tile0 = sp * 32 + w * TPW_;

    // prologue: async-load tile 0 into buffer 0
    tdm_load_tile(x + ((size_t)(c * T_ + tile0 * 16)) * E_ + h * D_, ut_byte0);
    MEMBAR();

    v2f acc2[16] = {};                   // per-lane column-sum partials

    for (int it = 0; it < TPW_; ++it) {
        // async-load next tile into the other buffer, then wait for current
        if (it + 1 < TPW_) {
            tdm_load_tile(x + ((size_t)(c * T_ + (tile0 + it + 1) * 16)) * E_ + h * D_,
                          ut_byte0 + ((it + 1) & 1) * (16 * 68 * 4));
            MEMBAR();
            __builtin_amdgcn_s_wait_tensorcnt(1);
        } else {
            __builtin_amdgcn_s_wait_tensorcnt(0);
        }
        MEMBAR();

        const float* ub = &ut[(it & 1) * (16 * 68)];
        const int t0 = (tile0 + it) * 16;

        // A fragments of U from LDS: lane=row, cols 4k+2*half (+1), b64 loads
        v2f af[16];
        #pragma unroll
        for (int k = 0; k < 16; ++k)
            af[k] = *(const v2f*)&ub[row * 68 + 4 * k + 2 * half];

        #pragma unroll
        for (int k = 0; k < 16; ++k) acc2[k] += af[k];

        // Q = U * M : four independent accumulator chains; B frags streamed
        // from LDS per k-step (merged ds_load_2addr_b64, keeps VGPRs < 256)
        v8f q0 = {}, q1 = {}, q2 = {}, q3 = {};
        #pragma unroll
        for (int k = 0; k < 16; ++k) {
            const v2f b0 = *(const v2f*)&lmt[( 0 + row) * 68 + 4 * k + 2 * half];
            const v2f b1 = *(const v2f*)&lmt[(16 + row) * 68 + 4 * k + 2 * half];
            const v2f b2 = *(const v2f*)&lmt[(32 + row) * 68 + 4 * k + 2 * half];
            const v2f b3 = *(const v2f*)&lmt[(48 + row) * 68 + 4 * k + 2 * half];
            q0 = __builtin_amdgcn_wmma_f32_16x16x4_f32(false, af[k], false, b0, (short)0, q0, false, false);
            q1 = __builtin_amdgcn_wmma_f32_16x16x4_f32(false, af[k], false, b1, (short)0, q1, false, false);
            q2 = __builtin_amdgcn_wmma_f32_16x16x4_f32(false, af[k], false, b2, (short)0, q2, false, false);
            q3 = __builtin_amdgcn_wmma_f32_16x16x4_f32(false, af[k], false, b3, (short)0, q3, false, false);
        }

        // Relayout Q: C-layout -> row-major LDS -> A-layout fragments
        #pragma unroll
        for (int r = 0; r < 8; ++r) {
            const int m = r + 8 * half;
            qs[m * 68 +  0 + row] = q0[r];
            qs[m * 68 + 16 + row] = q1[r];
            qs[m * 68 + 32 + row] = q2[r];
            qs[m * 68 + 48 + row] = q3[r];
        }

        // P = Q * U^T ; B fragments of U^T are identical to A fragments of U
        v8f p8 = {};
        #pragma unroll
        for (int k = 0; k < 16; ++k) {
            v2f qa = *(const v2f*)&qs[row * 68 + 4 * k + 2 * half];
            p8 = __builtin_amdgcn_wmma_f32_16x16x4_f32(false, qa, false, af[k], (short)0, p8, false, false);
        }

        // diag(P): element (m,m) lives in vgpr m&7, lane (m<8 ? m : m+16)
        const int src = ((l & 8) ? (l + 16) : l) & 31;
        float si = 0.f;
        #pragma unroll
        for (int r = 0; r < 8; ++r) {
            float tv = __shfl(p8[r], src, 32);
            if ((l & 7) == r) si = tv;
        }
        if (l < 16)
            sbuf[((size_t)(c * T_ + t0 + l)) * H_ + h] = si * (1.0f / 32.0f); // 1/sqrt(E)
    }

    // reduce column sums across the 16 rows held in each lane-half
    #pragma unroll
    for (int m = 1; m <= 8; m <<= 1)
        #pragma unroll
        for (int k = 0; k < 16; ++k) {
            acc2[k].x += __shfl_xor(acc2[k].x, m, 32);
            acc2[k].y += __shfl_xor(acc2[k].y, m, 32);
        }
    if (row == 0) {   // lanes 0 and 16 hold the half-sums
        #pragma unroll
        for (int k = 0; k < 16; ++k) {
            pxs[(size_t)gw * 64 + 4 * k + 2 * half + 0] = acc2[k].x;
            pxs[(size_t)gw * 64 + 4 * k + 2 * half + 1] = acc2[k].y;
        }
    }
}

// ---------------- kernel 2: xsum reduce + vsum = xsum * Wv^T ----------------
__global__ __launch_bounds__(64)
void k2_vsum(const float* __restrict__ Wv, const float* __restrict__ pxs,
             float* __restrict__ vsum) {
    __shared__ float xs[64];
    const int bid = blockIdx.x;        // c*H + h
    const int d = threadIdx.x;
    float a = 0.f;
    for (int i = 0; i < 64; ++i)       // 64 waves contributed to this (c,h)
        a += pxs[((size_t)bid * 64 + i) * 64 + d];
    xs[d] = a;
    __syncthreads();
    float v = 0.f;
    #pragma unroll 8
    for (int d2 = 0; d2 < 64; ++d2) v += Wv[d * 64 + d2] * xs[d2];
    vsum[bid * 64 + d] = v;
}

// ---------------- kernel 3: softmax over C + broadcast multiply -------------
__global__ __launch_bounds__(256)
void k3_out(const float* __restrict__ sbuf, const float* __restrict__ vsum,
            float* __restrict__ out) {
    __shared__ float pr[C_ * H_];
    const int t = blockIdx.x;
    const int tid = threadIdx.x;
    if (tid < H_) {
        const int h = tid;
        float sv[C_];
        float mx = -3.4e38f;
        #pragma unroll
        for (int c = 0; c < C_; ++c) {
            sv[c] = sbuf[((size_t)(c * T_ + t)) * H_ + h];
            mx = fmaxf(mx, sv[c]);
        }
        float sum = 0.f;
        #pragma unroll
        for (int c = 0; c < C_; ++c) { sv[c] = __expf(sv[c] - mx); sum += sv[c]; }
        const float inv = 1.f / sum;
        #pragma unroll
        for (int c = 0; c < C_; ++c) pr[c * H_ + h] = sv[c] * inv;
    }
    __syncthreads();
    const int c = tid >> 5;
    const int base = (tid & 31) * 4;
    #pragma unroll
    for (int j = 0; j < 8; ++j) {
        const int e = base + j * 128;
        const int h = e >> 6;
        const v4f vs = *(const v4f*)&vsum[(c * H_ + h) * 64 + (e & 63)];
        const float p = pr[c * H_ + h];
        v4f o = p * vs;
        __builtin_nontemporal_store(o, (v4f*)&out[((size_t)(c * T_ + t)) * E_ + e]);
    }
}

extern "C" void kernel_launch(void* const* d_in, const int* in_sizes, int n_in,
                              void* d_out, int out_size, void* d_ws, size_t ws_size,
                              hipStream_t stream) {
    const float* x  = (const float*)d_in[0];
    const float* Wq = (const float*)d_in[1];
    const float* Wk = (const float*)d_in[2];
    const float* Wv = (const float*)d_in[3];
    float* out = (float*)d_out;

    float* ws   = (float*)d_ws;
    float* mt   = ws;                                   // 4096 f32
    float* sbuf = ws + 4096;                            // C*T*H = 524288 f32
    float* pxs  = sbuf + (size_t)C_ * T_ * H_;          // 8192 waves * 64 f32
    float* vsum = pxs + (size_t)(C_ * H_ * SPLIT_ * 8) * 64; // C*H*64 f32

    hipLaunchKernelGGL(k0_prep,  dim3(1),                 dim3(256), 0, stream, Wq, Wk, mt);
    hipLaunchKernelGGL(k1_scores,dim3(C_ * H_ * SPLIT_),  dim3(256), 0, stream, x, mt, sbuf, pxs);
    hipLaunchKernelGGL(k2_vsum,  dim3(C_ * H_),           dim3(64),  0, stream, Wv, pxs, vsum);
    hipLaunchKernelGGL(k3_out,   dim3(T_),                dim3(256), 0, stream, sbuf, vsum, out);
}